// Quan2d_37512244363617
// MI455X (gfx1250) — compile-verified
//
#include <hip/hip_runtime.h>
#include <hip/hip_bf16.h>

// Quantum patch circuit, restructured as:
//   per patch: u = Kron(RY angles) applied to |0000>  (8 trig ops)
//   w = V * u   -> batched as 16x16x16 GEMM via 4x V_WMMA_F32_16X16X4_F32
//   ev[q] = sum_a sign_q(a) * w[a]^2
// V is a fixed 16x16 matrix from `weights`, built once on-device (d_ws).

typedef float v2f __attribute__((ext_vector_type(2)));
typedef float v8f __attribute__((ext_vector_type(8)));

// ---------------- setup kernel: build V (16x16, row-major V[a][i]) -------------

__device__ __forceinline__ void apply_ry(float v[16], int mask, float c, float s) {
#pragma unroll
  for (int i = 0; i < 16; ++i) {
    if ((i & mask) == 0) {
      int j = i | mask;
      float a = v[i], b = v[j];
      v[i] = c * a - s * b;   // new0 = c*old0 - s*old1
      v[j] = s * a + c * b;   // new1 = s*old0 + c*old1
    }
  }
}

__device__ __forceinline__ void apply_cnot(float v[16], int mc, int mt) {
#pragma unroll
  for (int i = 0; i < 16; ++i) {
    if ((i & mc) && !(i & mt)) {
      int j = i | mt;
      float t = v[i]; v[i] = v[j]; v[j] = t;
    }
  }
}

__global__ void build_V_kernel(const float* __restrict__ wts, float* __restrict__ V) {
  int t = threadIdx.x;
  if (t >= 16) return;
  float w[16];
#pragma unroll
  for (int i = 0; i < 16; ++i) w[i] = wts[i];
  float v[16];
#pragma unroll
  for (int i = 0; i < 16; ++i) v[i] = (i == t) ? 1.0f : 0.0f;
#pragma unroll
  for (int d = 0; d < 2; ++d) {
#pragma unroll
    for (int q = 0; q < 4; ++q) {
      float th = 0.5f * w[d * 8 + q];
      apply_ry(v, 1 << (3 - q), __cosf(th), __sinf(th));
    }
#pragma unroll
    for (int q = 0; q < 4; ++q)
      apply_cnot(v, 1 << (3 - q), 1 << (3 - ((q + 1) & 3)));
#pragma unroll
    for (int q = 0; q < 4; ++q) {
      float th = 0.5f * w[d * 8 + 4 + q];
      apply_ry(v, 1 << (3 - q), __cosf(th), __sinf(th));
    }
  }
  // column t of V = circuit applied to e_t ; store row-major V[a][t]
#pragma unroll
  for (int a = 0; a < 16; ++a) V[a * 16 + t] = v[a];
}

// ---------------- main kernel -------------------------------------------------

#define WAVES_PER_BLOCK 8

__global__ __launch_bounds__(256) void circuit_kernel(
    const float* __restrict__ x, const float* __restrict__ V,
    float* __restrict__ out, int numTiles) {
  // One wave handles a tile of 16 patches per loop iteration.
  __shared__ float lds[WAVES_PER_BLOCK * 256];
  const int lane = threadIdx.x & 31;
  const int wave = threadIdx.x >> 5;
  float* wlds = lds + wave * 256;
  const int row = lane & 15;   // patch-in-tile for A, state index n for B/D
  const int hi  = lane >> 4;   // K half-select

  // B fragments (V^T chunks): lane l, vgpr j -> B[k][n] = V[n][k],
  // n = lane&15, k = 4*chunk + 2*hi + j. Constant for the whole kernel.
  const float* vr = V + row * 16 + 2 * hi;
  v2f b0 = *(const v2f*)(vr + 0);
  v2f b1 = *(const v2f*)(vr + 4);
  v2f b2 = *(const v2f*)(vr + 8);
  v2f b3 = *(const v2f*)(vr + 12);

  // ---- loop-invariant output bookkeeping -------------------------------------
  // Lane produces outputs o = lane and o = lane+32; o = p*4 + q, so
  // q = lane&3 is the same for both and p differs by 8.
  // sign_q(a) < 0 iff bit (3-q) of a is set; packed mask table:
  //   q=0 -> 0xFF00, q=1 -> 0xF0F0, q=2 -> 0xCCCC, q=3 -> 0xAAAA
  const unsigned q = (unsigned)(lane & 3);
  const unsigned mask =
      (unsigned)(0xAAAACCCCF0F0FF00ull >> (q << 4)) & 0xFFFFu;
  int sb[16];                  // per-element sign-bit XOR masks (registers)
#pragma unroll
  for (int a = 0; a < 16; ++a) sb[a] = (int)(((mask >> a) & 1u) << 31);
  const float* wrow0 = wlds + (lane >> 2) * 16;        // patch p
  const float* wrow1 = wrow0 + 8 * 16;                 // patch p+8

  const int tileStride = gridDim.x * WAVES_PER_BLOCK;
  for (int tile = blockIdx.x * WAVES_PER_BLOCK + wave; tile < numTiles;
       tile += tileStride) {
    // ---- per-patch encoding state u (lanes l and l+16 redundantly own patch row)
    int P   = tile * 16 + row;        // global patch id
    int b   = P >> 10;                // 1024 patches per 64x64 image
    int rem = P & 1023;
    int ph  = rem >> 5, pw = rem & 31;
    const float* px = x + b * 4096 + ph * 128 + pw * 2;
    v2f r0 = *(const v2f*)(px);        // angles for qubits 0,1
    v2f r1 = *(const v2f*)(px + 64);   // angles for qubits 2,3

    float c0 = __cosf(0.5f * r0.x), s0 = __sinf(0.5f * r0.x);
    float c1 = __cosf(0.5f * r0.y), s1 = __sinf(0.5f * r0.y);
    float c2 = __cosf(0.5f * r1.x), s2 = __sinf(0.5f * r1.x);
    float c3 = __cosf(0.5f * r1.y), s3 = __sinf(0.5f * r1.y);

    // A fragment: lane needs u[k], k = 4*chunk + 2*hi + j.
    // k bits: [chunk(2) | hi(1) | j(1)] = qubit bits (q0 q1 q2 q3).
    float t2 = hi ? s2 : c2;
    float m0 = t2 * c3, m1 = t2 * s3;
    float p00 = c0 * c1, p01 = c0 * s1, p10 = s0 * c1, p11 = s0 * s1;
    v2f a0 = { p00 * m0, p00 * m1 };
    v2f a1 = { p01 * m0, p01 * m1 };
    v2f a2 = { p10 * m0, p10 * m1 };
    v2f a3 = { p11 * m0, p11 * m1 };

    // ---- W_tile(16 patches x 16 states) = U * V^T via 4 chained f32 WMMAs
    v8f c = {0.f, 0.f, 0.f, 0.f, 0.f, 0.f, 0.f, 0.f};
    c = __builtin_amdgcn_wmma_f32_16x16x4_f32(false, a0, false, b0, (short)0, c, false, false);
    c = __builtin_amdgcn_wmma_f32_16x16x4_f32(false, a1, false, b1, (short)0, c, false, false);
    c = __builtin_amdgcn_wmma_f32_16x16x4_f32(false, a2, false, b2, (short)0, c, false, false);
    c = __builtin_amdgcn_wmma_f32_16x16x4_f32(false, a3, false, b3, (short)0, c, false, false);

    // ---- D layout: lane l, vgpr r -> W[p = r + 8*hi][a = row]; stage via LDS
#pragma unroll
    for (int r = 0; r < 8; ++r) wlds[(r + 8 * hi) * 16 + row] = c[r];
    // wave-synchronous LDS: same-wave DS ops are in-order; just drain DScnt
    asm volatile("s_wait_dscnt 0" ::: "memory");

    // ---- ev[p][q] = sum_a sign_q(a) * W[p][a]^2 ; branchless xor-sign FMA
    float acc0 = 0.0f, acc1 = 0.0f;
#pragma unroll
    for (int a = 0; a < 16; ++a) {
      float v0 = wrow0[a];
      float v1 = wrow1[a];
      float sv0 = __int_as_float(__float_as_int(v0) ^ sb[a]);
      float sv1 = __int_as_float(__float_as_int(v1) ^ sb[a]);
      acc0 = fmaf(sv0, v0, acc0);
      acc1 = fmaf(sv1, v1, acc1);
    }
    // (no second dscnt drain needed: same-wave LDS ops stay in order, so next
    //  tile's ds_store cannot bypass these ds_loads in the LDS pipeline)

    float* obase = out + tile * 64;   // out flat index = P*4 + q = tile*64 + o
    obase[lane]      = acc0;
    obase[lane + 32] = acc1;
  }
}

// ---------------- launch ------------------------------------------------------

extern "C" void kernel_launch(void* const* d_in, const int* in_sizes, int n_in,
                              void* d_out, int out_size, void* d_ws, size_t ws_size,
                              hipStream_t stream) {
  const float* x   = (const float*)d_in[0];   // (512,1,64,64) f32
  const float* wts = (const float*)d_in[1];   // (2,8) f32
  float* out = (float*)d_out;                 // (512,4096) f32
  float* V   = (float*)d_ws;                  // 16x16 f32 scratch

  int numPatches = in_sizes[0] / 4;           // 524288
  int numTiles   = numPatches / 16;           // 32768

  build_V_kernel<<<1, 32, 0, stream>>>(wts, V);

  int blocks = (numTiles + WAVES_PER_BLOCK - 1) / WAVES_PER_BLOCK;
  if (blocks > 1024) blocks = 1024;
  circuit_kernel<<<blocks, 256, 0, stream>>>(x, V, out, numTiles);
}